// cVAE_29274497089632
// MI455X (gfx1250) — compile-verified
//
#include <hip/hip_runtime.h>
#include <math.h>

// ---------------------------------------------------------------------------
// cVAE forward: bidirectional LSTM (T=20,H=10,IN=2) + attention GEMV
// (20x200 via chained V_WMMA_F32_16X16X4_F32, K split across wave pairs)
// + pooled MLP + relaxed-one-hot sample.
// Latency-bound -> single workgroup (128 threads = 4 wave32s), all state in
// LDS, fwd/bwd LSTMs on disjoint waves, att_W prefetched behind the scan.
// ---------------------------------------------------------------------------

typedef __attribute__((ext_vector_type(2))) float v2f;
typedef __attribute__((ext_vector_type(8))) float v8f;

static constexpr int kT  = 20;   // seq_len
static constexpr int kH  = 10;   // hidden
static constexpr int kIN = 2;    // input size
static constexpr int kNE = 8;    // n_embedding
static constexpr float kInvTemp = 5.0f;  // 1/0.2

__device__ __forceinline__ float sigm(float x) { return 1.0f / (1.0f + __expf(-x)); }

__global__ __launch_bounds__(128, 1)
void cVAE_fused_kernel(const float* __restrict__ x,
                       const float* __restrict__ W_ih_f, const float* __restrict__ W_hh_f,
                       const float* __restrict__ b_ih_f, const float* __restrict__ b_hh_f,
                       const float* __restrict__ W_ih_b, const float* __restrict__ W_hh_b,
                       const float* __restrict__ b_ih_b, const float* __restrict__ b_hh_b,
                       const float* __restrict__ att_W,  const float* __restrict__ att_b,
                       const float* __restrict__ mlp_W,  const float* __restrict__ mlp_b,
                       const float* __restrict__ gumbel,
                       float* __restrict__ out)
{
    __shared__ float s_x[kT * kIN];              // staged inputs
    __shared__ alignas(16) float s_h[2][12];     // current hidden (padded for b128)
    __shared__ float s_gates[2][4 * kH];         // gate pre-activations
    __shared__ float s_hf[kT * kH];              // forward hidden states
    __shared__ float s_hb[kT * kH];              // backward hidden states (time-aligned)
    __shared__ alignas(16) float s_out[kT * kH]; // pair-summed features (len 200)
    __shared__ float s_ypart[2][kT];             // K-split partial attention logits
    __shared__ float s_y[kT];                    // tanh(attention logits)
    __shared__ float s_e[kT];                    // attention exp()
    __shared__ float s_pool[kH];
    __shared__ float s_z[kNE];
    __shared__ float s_e2[kNE];
    __shared__ float s_red[4];

    const int tid = threadIdx.x;

    // ---- prefetch att_W (16 KB = 125 cachelines) behind the serial LSTM ----
    {
        const int off = tid * 32;                // one 128B line per thread
        if (off < kT * kT * kH) __builtin_prefetch(att_W + off, 0, 3);
    }

    // ---- role assignment for the LSTM phase -------------------------------
    // waves 0-1 (tid 0..63): forward direction, waves 2-3 (tid 64..127): backward.
    const int dir    = (tid < 64) ? 0 : 1;
    const int g      = tid & 63;                 // gate row within direction
    const bool gateT = (g < 4 * kH);             // 40 gate rows per direction

    // Per-thread weight preload: W_ih row (2), fused bias, W_hh row (10).
    float wih0 = 0.f, wih1 = 0.f, bsum = 0.f;
    float wh[kH];
#pragma unroll
    for (int j = 0; j < kH; ++j) wh[j] = 0.f;
    if (gateT) {
        const float* Wih = dir ? W_ih_b : W_ih_f;
        const float* Whh = dir ? W_hh_b : W_hh_f;
        const float* bih = dir ? b_ih_b : b_ih_f;
        const float* bhh = dir ? b_hh_b : b_hh_f;
        wih0 = Wih[g * kIN + 0];
        wih1 = Wih[g * kIN + 1];
        bsum = bih[g] + bhh[g];
#pragma unroll
        for (int j = 0; j < kH; ++j) wh[j] = Whh[g * kH + j];
    }

    // Stage inputs / init hidden state (incl. padding lanes).
    if (tid < kT * kIN) s_x[tid] = x[tid];
    if (tid < 24)       s_h[tid / 12][tid % 12] = 0.f;

    float cc = 0.f;  // cell state, lives in threads g<10 of each direction

    // ---- sequential LSTM scan (both directions concurrently) --------------
    for (int t = 0; t < kT; ++t) {
        __syncthreads();  // h (and, at t=0, s_x) ready; previous gates consumed
        if (gateT) {
            const int tt = dir ? (kT - 1 - t) : t;
            const float4 h0 = *(const float4*)&s_h[dir][0];   // ds_load_b128
            const float4 h1 = *(const float4*)&s_h[dir][4];   // ds_load_b128
            const float2 h2 = *(const float2*)&s_h[dir][8];   // ds_load_b64
            float acc = bsum + wih0 * s_x[tt * kIN] + wih1 * s_x[tt * kIN + 1];
            acc += wh[0] * h0.x + wh[1] * h0.y + wh[2] * h0.z + wh[3] * h0.w;
            acc += wh[4] * h1.x + wh[5] * h1.y + wh[6] * h1.z + wh[7] * h1.w;
            acc += wh[8] * h2.x + wh[9] * h2.y;
            s_gates[dir][g] = acc;
        }
        __syncthreads();
        if (gateT && g < kH) {  // cell update, PyTorch gate order i,f,g,o
            const float gi = sigm(s_gates[dir][g]);
            const float gf = sigm(s_gates[dir][kH + g]);
            const float gg = tanhf(s_gates[dir][2 * kH + g]);
            const float go = sigm(s_gates[dir][3 * kH + g]);
            cc = gf * cc + gi * gg;
            const float hh = go * tanhf(cc);
            s_h[dir][g] = hh;
            if (dir == 0) s_hf[t * kH + g] = hh;
            else          s_hb[(kT - 1 - t) * kH + g] = hh;
        }
    }
    __syncthreads();

    // ---- concat + reshape(T,H,2).sum(-1): pair adjacent channels ----------
    for (int idx = tid; idx < kT * kH; idx += 128) {
        const int t = idx / kH, j = idx % kH;
        s_out[idx] = (j < 5)
            ? (s_hf[t * kH + 2 * j]        + s_hf[t * kH + 2 * j + 1])
            : (s_hb[t * kH + (2 * j - 10)] + s_hb[t * kH + (2 * j - 9)]);
    }
    __syncthreads();

    // ---- attention logits: y[20] = att_W(20x200) @ s_out(200) + att_b -----
    // D = A(16x4) x B(4x16) + C, chained.  EVERY row of A carries the vector
    // segment (all D rows equal -> unconditional broadcast ds_load_b64, no
    // masking; we read only row M=0 = D vgpr0 lanes 0-15).  B holds the
    // transposed att_W tile; out-of-range rows clamp their address to row 19
    // and the garbage columns are simply never stored.  K=200 is split in two
    // halves across wave pairs: 25 dependent WMMAs per wave instead of 50.
    //   wave 0: rows 0-15, K 0..99     wave 1: rows 0-15, K 100..199
    //   wave 2: rows 16-19, K 0..99    wave 3: rows 16-19, K 100..199
    {
        const int lane    = tid & 31;
        const int waveId  = tid >> 5;
        const int half    = waveId & 1;
        const int rowbase = (waveId >> 1) * 16;
        const int n       = lane & 15;
        const int row     = rowbase + n;
        const int rowc    = (row < kT) ? row : (kT - 1);     // address clamp
        const int koff    = (lane >> 4) * 2;                 // K sub-offset per half-wave
        const float* aptr = s_out + half * 100 + koff;       // LDS, broadcast per half-wave
        const float* bptr = att_W + rowc * (kT * kH) + half * 100 + koff;
        v8f c = {0.f, 0.f, 0.f, 0.f, 0.f, 0.f, 0.f, 0.f};
#pragma unroll
        for (int kt = 0; kt < 25; ++kt) {
            const v2f a = *(const v2f*)(aptr + kt * 4);      // ds_load_b64
            const v2f b = *(const v2f*)(bptr + kt * 4);      // global_load_b64 (prefetched)
            c = __builtin_amdgcn_wmma_f32_16x16x4_f32(
                    false, a, false, b, (short)0, c, false, false);
        }
        if (lane < 16 && row < kT) s_ypart[half][row] = c[0];
    }
    __syncthreads();

    // ---- att = softmax(tanh(y)) -------------------------------------------
    if (tid < kT) s_y[tid] = tanhf(s_ypart[0][tid] + s_ypart[1][tid] + att_b[tid]);
    __syncthreads();
    if (tid == 0) {
        float m = s_y[0];
        for (int t = 1; t < kT; ++t) m = fmaxf(m, s_y[t]);
        s_red[0] = m;
    }
    __syncthreads();
    if (tid < kT) s_e[tid] = __expf(s_y[tid] - s_red[0]);
    __syncthreads();
    if (tid == 0) {
        float s = 0.f;
        for (int t = 0; t < kT; ++t) s += s_e[t];
        s_red[1] = s;
    }
    __syncthreads();

    // ---- pooled[j] = sum_t out[t][j] * att[t] -----------------------------
    if (tid < kH) {
        float p = 0.f;
        for (int t = 0; t < kT; ++t) p += s_out[t * kH + tid] * s_e[t];
        s_pool[tid] = p / s_red[1];
    }
    __syncthreads();

    // ---- MLP logits + gumbel; softmax((log softmax(z)+g)/tau)
    //      == softmax((z+g)/tau): the logsumexp shift cancels. ---------------
    if (tid < kNE) {
        float z = mlp_b[tid];
#pragma unroll
        for (int j = 0; j < kH; ++j) z += mlp_W[tid * kH + j] * s_pool[j];
        s_z[tid] = (z + gumbel[tid]) * kInvTemp;
    }
    __syncthreads();
    if (tid == 0) {
        float m = s_z[0];
        for (int e = 1; e < kNE; ++e) m = fmaxf(m, s_z[e]);
        s_red[2] = m;
    }
    __syncthreads();
    if (tid < kNE) s_e2[tid] = __expf(s_z[tid] - s_red[2]);
    __syncthreads();
    if (tid == 0) {
        float s = 0.f;
        for (int e = 0; e < kNE; ++e) s += s_e2[e];
        s_red[3] = s;
    }
    __syncthreads();
    if (tid < kNE) out[tid] = s_e2[tid] / s_red[3];
}

extern "C" void kernel_launch(void* const* d_in, const int* in_sizes, int n_in,
                              void* d_out, int out_size, void* d_ws, size_t ws_size,
                              hipStream_t stream) {
    (void)in_sizes; (void)n_in; (void)out_size; (void)d_ws; (void)ws_size;
    const float* x      = (const float*)d_in[0];
    const float* W_ih_f = (const float*)d_in[1];
    const float* W_hh_f = (const float*)d_in[2];
    const float* b_ih_f = (const float*)d_in[3];
    const float* b_hh_f = (const float*)d_in[4];
    const float* W_ih_b = (const float*)d_in[5];
    const float* W_hh_b = (const float*)d_in[6];
    const float* b_ih_b = (const float*)d_in[7];
    const float* b_hh_b = (const float*)d_in[8];
    const float* att_W  = (const float*)d_in[9];
    const float* att_b  = (const float*)d_in[10];
    const float* mlp_W  = (const float*)d_in[11];
    const float* mlp_b  = (const float*)d_in[12];
    const float* gum    = (const float*)d_in[13];
    float* outp = (float*)d_out;

    cVAE_fused_kernel<<<1, 128, 0, stream>>>(
        x, W_ih_f, W_hh_f, b_ih_f, b_hh_f,
        W_ih_b, W_hh_b, b_ih_b, b_hh_b,
        att_W, att_b, mlp_W, mlp_b, gum, outp);
}